// HybridLoss_40544491274621
// MI455X (gfx1250) — compile-verified
//
#include <hip/hip_runtime.h>
#include <cstdint>

// ---- constants from the reference ----
#define HEATMAP_W_C    1.0f
#define COORD_W_C      2.0f
#define OFFSET_SCALE_C 0.03f
#define OFFSET_REG_W_C 0.1f
#define WEIGHT_GAIN_C  5.0f
// exp(20*x) == exp2(x * 20/ln2)
#define LOG2E_TEMP_C   28.853900817779268f

constexpr int Bn     = 512;
constexpr int Hn     = 160;
constexpr int Wn     = 160;
constexpr int NELEM  = Hn * Wn;        // 25600 per image (100 KB fp32)
constexpr int TPB    = 256;            // 8 wave32 waves
constexpr int NVEC   = NELEM / 4;      // 6400 float4 chunks
constexpr int CHUNKS = NVEC / TPB;     // 25 chunks per lane

// 16-byte GCC vector type matching the builtin's parameter pointee type
typedef int v4i __attribute__((vector_size(16)));

// ---------------- CDNA5 async global->LDS copy (ASYNCcnt path) ----------------
__device__ __forceinline__ void async_copy_f4(const float* g, float* l) {
#if __has_builtin(__builtin_amdgcn_global_load_async_to_lds_b128)
    __builtin_amdgcn_global_load_async_to_lds_b128(
        (__attribute__((address_space(1))) v4i*)(g),
        (__attribute__((address_space(3))) v4i*)(l),
        0, 0);
#else
    // generic pointer into LDS: low 32 bits are the LDS byte address
    unsigned lds_off = (unsigned)(unsigned long long)(uintptr_t)l;
    asm volatile("global_load_async_to_lds_b128 %0, %1, off"
                 :: "v"(lds_off), "v"((unsigned long long)(uintptr_t)g)
                 : "memory");
#endif
}

__device__ __forceinline__ void async_wait0() {
#if __has_builtin(__builtin_amdgcn_s_wait_asynccnt)
    __builtin_amdgcn_s_wait_asynccnt(0);
#else
    asm volatile("s_wait_asynccnt 0" ::: "memory");
#endif
}

// ---------------- wave32 / block reductions ----------------
__device__ __forceinline__ float wave_sum(float v) {
#pragma unroll
    for (int o = 16; o > 0; o >>= 1) v += __shfl_xor(v, o, 32);
    return v;
}
__device__ __forceinline__ float wave_max(float v) {
#pragma unroll
    for (int o = 16; o > 0; o >>= 1) v = fmaxf(v, __shfl_xor(v, o, 32));
    return v;
}

template <int NWAVES>
__device__ __forceinline__ float block_sum(float v, float* scr) {
    v = wave_sum(v);
    const int lane = threadIdx.x & 31;
    const int wid  = threadIdx.x >> 5;
    __syncthreads();                       // protect scratch reuse
    if (lane == 0) scr[wid] = v;
    __syncthreads();
    float r = (lane < NWAVES) ? scr[lane] : 0.0f;
    return wave_sum(r);                    // all threads get the result
}

template <int NWAVES>
__device__ __forceinline__ float block_max(float v, float* scr) {
    v = wave_max(v);
    const int lane = threadIdx.x & 31;
    const int wid  = threadIdx.x >> 5;
    __syncthreads();
    if (lane == 0) scr[wid] = v;
    __syncthreads();
    float r = (lane < NWAVES) ? scr[lane] : -3.402823466e38f;
    return wave_max(r);
}

// ---------------- kernel 1: per-image fused reduction ----------------
__global__ __launch_bounds__(TPB)
void hybrid_loss_main(const float* __restrict__ pred,
                      const float* __restrict__ tgt,
                      const float* __restrict__ poff,
                      const float* __restrict__ tcoord,
                      float* __restrict__ ws_hm,
                      float* __restrict__ ws_coord,
                      float* __restrict__ ws_off) {
    __shared__ __align__(16) float s_data[NELEM];   // 100 KB of the 320 KB WGP LDS
    __shared__ float s_scr[TPB / 32];

    const int b   = blockIdx.x;
    const int tid = threadIdx.x;
    const float* gp = pred + (size_t)b * NELEM;
    const float* gt = tgt  + (size_t)b * NELEM;

    // Stage pred image into LDS via async DMA (no VGPR round trip).
    // Lane-consecutive 16B chunks -> fully coalesced 512B per wave instruction.
#pragma unroll
    for (int k = 0; k < CHUNKS; ++k) {
        const int i = tid + k * TPB;
        async_copy_f4(gp + 4 * i, &s_data[4 * i]);
    }
    async_wait0();
    __syncthreads();

    // ---- pass 1: weighted MSE partial + logit max (softmax stability) ----
    float lsum = 0.0f;
    float vmax = -3.402823466e38f;
#pragma unroll
    for (int k = 0; k < CHUNKS; ++k) {
        const int i = tid + k * TPB;
        const float4 p = *reinterpret_cast<const float4*>(&s_data[4 * i]);
        const float4 t = *reinterpret_cast<const float4*>(&gt[4 * i]);
        const float d0 = p.x - t.x, d1 = p.y - t.y, d2 = p.z - t.z, d3 = p.w - t.w;
        lsum += d0 * d0 * (1.0f + WEIGHT_GAIN_C * t.x);
        lsum += d1 * d1 * (1.0f + WEIGHT_GAIN_C * t.y);
        lsum += d2 * d2 * (1.0f + WEIGHT_GAIN_C * t.z);
        lsum += d3 * d3 * (1.0f + WEIGHT_GAIN_C * t.w);
        vmax = fmaxf(vmax, fmaxf(fmaxf(p.x, p.y), fmaxf(p.z, p.w)));
    }
    lsum = block_sum<TPB / 32>(lsum, s_scr);
    const float m = block_max<TPB / 32>(vmax, s_scr);

    // ---- pass 2: softmax sums + coordinate expectation (LDS re-read, free) ----
    float s = 0.0f, sx = 0.0f, sy = 0.0f;
#pragma unroll
    for (int k = 0; k < CHUNKS; ++k) {
        const int i = tid + k * TPB;
        const int j = 4 * i;
        const float fx = (float)(j % Wn);   // 160 % 4 == 0 -> same row per float4
        const float fy = (float)(j / Wn);
        const float4 p = *reinterpret_cast<const float4*>(&s_data[4 * i]);
        const float e0 = exp2f((p.x - m) * LOG2E_TEMP_C);
        const float e1 = exp2f((p.y - m) * LOG2E_TEMP_C);
        const float e2 = exp2f((p.z - m) * LOG2E_TEMP_C);
        const float e3 = exp2f((p.w - m) * LOG2E_TEMP_C);
        const float es = (e0 + e1) + (e2 + e3);
        s  += es;
        sx += e0 * fx + e1 * (fx + 1.0f) + e2 * (fx + 2.0f) + e3 * (fx + 3.0f);
        sy += es * fy;
    }
    s  = block_sum<TPB / 32>(s,  s_scr);
    sx = block_sum<TPB / 32>(sx, s_scr);
    sy = block_sum<TPB / 32>(sy, s_scr);

    if (tid == 0) {
        const float px = (sx / s) * (1.0f / (float)Wn);
        const float py = (sy / s) * (1.0f / (float)Hn);
        const float ox = poff[2 * b], oy = poff[2 * b + 1];
        const float fx = px + ox * OFFSET_SCALE_C;
        const float fy = py + oy * OFFSET_SCALE_C;
        const float dx = fx - tcoord[2 * b];
        const float dy = fy - tcoord[2 * b + 1];
        ws_hm[b]    = lsum;
        ws_coord[b] = dx * dx + dy * dy;
        ws_off[b]   = ox * ox + oy * oy;
    }
}

// ---------------- kernel 2: final deterministic reduction ----------------
__global__ __launch_bounds__(Bn)
void hybrid_loss_final(const float* __restrict__ ws_hm,
                       const float* __restrict__ ws_coord,
                       const float* __restrict__ ws_off,
                       float* __restrict__ out) {
    __shared__ float scr[Bn / 32];
    const int t = threadIdx.x;
    float h = ws_hm[t];
    float c = ws_coord[t];
    float o = ws_off[t];
    h = block_sum<Bn / 32>(h, scr);
    c = block_sum<Bn / 32>(c, scr);
    o = block_sum<Bn / 32>(o, scr);
    if (t == 0) {
        const float hm_loss    = h * (1.0f / ((float)Bn * (float)NELEM));
        const float coord_loss = c * (1.0f / (float)(Bn * 2));
        const float off_reg    = o * (1.0f / (float)(Bn * 2)) * OFFSET_REG_W_C;
        const float total = HEATMAP_W_C * hm_loss + COORD_W_C * coord_loss + off_reg;
        out[0] = total;
        out[1] = hm_loss;
        out[2] = coord_loss;
    }
}

// ---------------- host entry ----------------
extern "C" void kernel_launch(void* const* d_in, const int* in_sizes, int n_in,
                              void* d_out, int out_size, void* d_ws, size_t ws_size,
                              hipStream_t stream) {
    (void)in_sizes; (void)n_in; (void)out_size; (void)ws_size;
    const float* pred   = (const float*)d_in[0];   // (512,1,160,160)
    const float* tgt    = (const float*)d_in[1];   // (512,1,160,160)
    const float* poff   = (const float*)d_in[2];   // (512,2)
    const float* tcoord = (const float*)d_in[3];   // (512,2)
    // d_in[4] == heatmap_size (160) -> baked into constants

    float* ws       = (float*)d_ws;                // 3 * 512 floats, fully rewritten every call
    float* ws_hm    = ws;
    float* ws_coord = ws + Bn;
    float* ws_off   = ws + 2 * Bn;

    hybrid_loss_main<<<Bn, TPB, 0, stream>>>(pred, tgt, poff, tcoord,
                                             ws_hm, ws_coord, ws_off);
    hybrid_loss_final<<<1, Bn, 0, stream>>>(ws_hm, ws_coord, ws_off, (float*)d_out);
}